// DistanceBasedQueryScorer_20057497272860
// MI455X (gfx1250) — compile-verified
//
#include <hip/hip_runtime.h>
#include <hip/hip_bf16.h>

#define EPSF      1e-8f
#define NFREQ     64
#define NBINS     128
#define HEADDIM   128
#define NQUERIES  32768
#define QTILE     16      // queries per block (WMMA M)
#define NTHREADS  256     // 8 waves of 32

typedef __attribute__((ext_vector_type(2))) float v2f;
typedef __attribute__((ext_vector_type(8))) float v8f;
typedef __attribute__((ext_vector_type(4))) unsigned int u32x4;
typedef __attribute__((ext_vector_type(8))) int i32x8;
typedef __attribute__((ext_vector_type(4))) int i32x4;

#ifndef __has_builtin
#define __has_builtin(x) 0
#endif
#if __has_builtin(__builtin_amdgcn_tensor_load_to_lds) && __has_builtin(__builtin_amdgcn_s_wait_tensorcnt)
#define HAVE_TDM 1
#else
#define HAVE_TDM 0
#endif

// ---------------------------------------------------------------------------
// Prep kernel: materialize PW[f][b] = (P_real, P_imag, -softplus(w_raw), 0)
// once per call instead of once per 2048 blocks (kills redundant exp/log work).
// ---------------------------------------------------------------------------
__global__ __launch_bounds__(256)
void prep_pw_kernel(const float* __restrict__ probes,
                    const float* __restrict__ w_raw,
                    float4* __restrict__ pw_pre)
{
    int e = blockIdx.x * 256 + threadIdx.x;          // 8192 total
    if (e >= NFREQ * NBINS) return;
    int b = e & 127, f = e >> 7;                     // flat = f*128 + b
    float pr = probes[(size_t)b * HEADDIM + f];
    float pi = probes[(size_t)b * HEADDIM + NFREQ + f];
    float w  = -log1pf(__expf(w_raw[(size_t)b * NFREQ + f]));
    pw_pre[e] = make_float4(pr, pi, w, 0.0f);
}

// ---------------------------------------------------------------------------
// Main kernel. PRE=true: stage precomputed PW from global via the Tensor Data
// Mover (single 128KB 2D descriptor, one wave issues, TENSORcnt tracked).
// PRE=false: self-contained (recompute softplus in-kernel).
// ---------------------------------------------------------------------------
template <bool PRE>
__global__ __launch_bounds__(NTHREADS)
void DistanceBasedQueryScorer_kernel(const float* __restrict__ Q,
                                     const float* __restrict__ probes,   // [NBINS][HEADDIM]
                                     const float* __restrict__ w_raw,    // [NBINS][NFREQ]
                                     const float* __restrict__ w_mag,    // [NBINS][NFREQ]
                                     const float* __restrict__ bias,     // [NBINS]
                                     float* __restrict__ out,            // [NQUERIES][NBINS]
                                     const float4* __restrict__ pw_pre)  // [NFREQ*NBINS] or null
{
    __shared__ float4 PW[NFREQ][NBINS];       // (P_real, P_imag, -softplus(w), 0), transposed [f][b]
    __shared__ float  Qtile[QTILE][HEADDIM];
    __shared__ float  Qn_r[QTILE][NFREQ];
    __shared__ float  Qn_i[QTILE][NFREQ];
    __shared__ float2 Qmag2[NFREQ / 2][QTILE]; // paired magnitudes -> one ds_load_b64 per A fragment
    __shared__ float  dist[QTILE][NBINS];
    __shared__ float  inv_n[QTILE];

    const int tid   = threadIdx.x;
    const int qbase = blockIdx.x * QTILE;

    // 1) Load 16 Q rows, coalesced
    for (int e = tid; e < QTILE * HEADDIM; e += NTHREADS) {
        int q = e >> 7, c = e & 127;
        Qtile[q][c] = Q[(size_t)(qbase + q) * HEADDIM + c];
    }

    // 2) Stage PW into LDS
    if constexpr (PRE) {
#if HAVE_TDM
        if (tid < 32) {   // one wave issues the TDM DMA; EXEC ignored by TDM
            unsigned int lds_off = (unsigned int)(uintptr_t)(&PW[0][0]);
            unsigned long long ga = (unsigned long long)(uintptr_t)pw_pre;
            // D# group 0: count=1 valid user descriptor, lds_addr, global_addr, type=2
            u32x4 g0;
            g0.x = 1u;
            g0.y = lds_off;
            g0.z = (unsigned int)ga;
            g0.w = (unsigned int)((ga >> 32) & 0x1FFFFFFull) | (2u << 30);
            // D# group 1: 1D-as-2D copy, data_size=8B, 16384 elements = 128KB
            i32x8 g1;
            g1[0] = (int)(3u << 16);        // workgroup_mask=0, data_size=3 (8B)
            g1[1] = (int)(0x4000u << 16);   // tensor_dim0 = 16384 (low16 at bits 63:48)
            g1[2] = (int)(1u << 16);        // tensor_dim0 hi=0; tensor_dim1 = 1
            g1[3] = (int)(0x4000u << 16);   // tensor_dim1 hi=0; tile_dim0 = 16384
            g1[4] = 1;                      // tile_dim1 = 1; tile_dim2 = 0
            g1[5] = 16384;                  // tensor_dim0_stride
            g1[6] = 0;
            g1[7] = 0;
            i32x4 gz4 = {0, 0, 0, 0};       // groups 2/3 unused (<=2D)
            i32x8 gz8 = {0, 0, 0, 0, 0, 0, 0, 0};
            __builtin_amdgcn_tensor_load_to_lds(g0, g1, gz4, gz4, gz8, 0);
            __builtin_amdgcn_s_wait_tensorcnt(0);
        }
#else
        {
            float4* pwf = &PW[0][0];
            for (int e = tid; e < NFREQ * NBINS; e += NTHREADS) pwf[e] = pw_pre[e];
        }
#endif
    } else {
        for (int e = tid; e < NFREQ * NBINS; e += NTHREADS) {
            int b = e & 127, f = e >> 7;
            float pr = probes[(size_t)b * HEADDIM + f];
            float pi = probes[(size_t)b * HEADDIM + NFREQ + f];
            float w  = -log1pf(__expf(w_raw[(size_t)b * NFREQ + f]));
            PW[f][b] = make_float4(pr, pi, w, 0.0f);
        }
    }
    __syncthreads();

    // 3) Per-query inverse norms
    if (tid < QTILE) {
        float s = 0.0f;
        for (int k = 0; k < HEADDIM; ++k) { float v = Qtile[tid][k]; s = fmaf(v, v, s); }
        inv_n[tid] = 1.0f / (sqrtf(s) + EPSF);
    }
    __syncthreads();

    // 4) Normalized components + paired magnitudes (512 iters, 2 freqs each)
    for (int e = tid; e < QTILE * (NFREQ / 2); e += NTHREADS) {
        int q = e >> 5, j = e & 31;
        float inv = inv_n[q];
        int f0 = 2 * j;
        float qr0 = Qtile[q][f0]             * inv;
        float qi0 = Qtile[q][NFREQ + f0]     * inv;
        float qr1 = Qtile[q][f0 + 1]         * inv;
        float qi1 = Qtile[q][NFREQ + f0 + 1] * inv;
        Qn_r[q][f0]     = qr0;  Qn_i[q][f0]     = qi0;
        Qn_r[q][f0 + 1] = qr1;  Qn_i[q][f0 + 1] = qi1;
        Qmag2[j][q] = make_float2(sqrtf(fmaf(qr0, qr0, fmaf(qi0, qi0, EPSF))),
                                  sqrtf(fmaf(qr1, qr1, fmaf(qi1, qi1, EPSF))));
    }
    __syncthreads();

    // 5) Distance term: each thread owns 8 (q,b) pairs; lanes in a wave share q
    for (int i = tid; i < QTILE * NBINS; i += NTHREADS) {
        int q = i >> 7, b = i & 127;
        float acc = 0.0f;
        #pragma unroll 4
        for (int f = 0; f < NFREQ; ++f) {
            float4 pw = PW[f][b];
            float er = pw.x - Qn_r[q][f];
            float ei = pw.y - Qn_i[q][f];
            float d  = sqrtf(fmaf(er, er, fmaf(ei, ei, EPSF)));
            acc = fmaf(d, pw.z, acc);
        }
        dist[q][b] = acc;
    }
    __syncthreads();

    // 6) Magnitude GEMM via chained f32 WMMA (16 steps of K=4); wave w owns 16 bins
    const int wave = tid >> 5;
    const int lane = tid & 31;
    const int b0   = wave * 16;
    const int mn   = lane & 15;
    const int hi   = lane >> 4;          // K pair {0,1} vs {2,3}
    const int bcol = b0 + mn;

    const float* wm_row = w_mag + (size_t)bcol * NFREQ;

    v8f c = {};
    #pragma unroll
    for (int kk = 0; kk < 16; ++kk) {
        int k0 = kk * 4 + hi * 2;
        float2 av = Qmag2[k0 >> 1][mn];  // one ds_load_b64, conflict-free
        v2f a, bfrag;
        a.x = av.x;  a.y = av.y;
        bfrag.x = wm_row[k0];
        bfrag.y = wm_row[k0 + 1];
        c = __builtin_amdgcn_wmma_f32_16x16x4_f32(
                false, a, false, bfrag, (short)0, c, false, false);
    }

    // 7) Fuse distance + magnitude + bias, coalesced stores
    float bv = bias[bcol];
    #pragma unroll
    for (int r = 0; r < 8; ++r) {
        int q = hi ? (r + 8) : r;
        float score = dist[q][bcol] + c[r] + bv;
        out[(size_t)(qbase + q) * NBINS + bcol] = score;
    }
}

extern "C" void kernel_launch(void* const* d_in, const int* in_sizes, int n_in,
                              void* d_out, int out_size, void* d_ws, size_t ws_size,
                              hipStream_t stream) {
    const float* Q    = (const float*)d_in[0];
    const float* P    = (const float*)d_in[1];
    const float* Wr   = (const float*)d_in[2];
    const float* Wm   = (const float*)d_in[3];
    const float* bias = (const float*)d_in[4];
    float* out = (float*)d_out;

    dim3 grid(NQUERIES / QTILE);
    dim3 block(NTHREADS);

    const size_t pw_bytes = sizeof(float4) * NFREQ * NBINS;   // 128 KB
    if (ws_size >= pw_bytes) {
        float4* pw_pre = (float4*)d_ws;
        prep_pw_kernel<<<dim3(32), dim3(256), 0, stream>>>(P, Wr, pw_pre);
        DistanceBasedQueryScorer_kernel<true><<<grid, block, 0, stream>>>(
            Q, P, Wr, Wm, bias, out, pw_pre);
    } else {
        DistanceBasedQueryScorer_kernel<false><<<grid, block, 0, stream>>>(
            Q, P, Wr, Wm, bias, out, nullptr);
    }
}